// GMMv1_12086037971142
// MI455X (gfx1250) — compile-verified
//
#include <hip/hip_runtime.h>
#include <hip/hip_bf16.h>
#include <math.h>

// GMM log-likelihood, N=8 channels, K=4 clusters.
// Strategy: prep kernel builds M = stacked L_k^{-1} (32x8), c = A_k*mu_k (32),
// const_k = log_w_k - log|L_k| - 0.5*N*log(2pi).
// Main kernel: Z[32 x 16voxels] = M * X via V_WMMA_F32_16X16X4_F32 (2 row
// tiles x 2 K-chunks), C accumulator pre-loaded with -c. Then per-lane
// sum-of-squares + cross-half shuffle for the 4-way logsumexp.

#define GMM_N 8
#define GMM_K 4
#define GMM_TRI 36
#define LOG_2PI 1.8378770664093453f

typedef float v2f __attribute__((ext_vector_type(2)));
typedef float v8f __attribute__((ext_vector_type(8)));

// workspace layout (floats):
//   [0   .. 255] Minv : 32 rows x 8 cols, row r=(k*8+n), Minv[r][c] = (L_k^{-1})[n][c]
//   [256 .. 287] cvec : cvec[k*8+n] = (L_k^{-1} mu_k)[n]
//   [288 .. 291] ck   : log_w_k - logdet_k - 0.5*N*log(2pi)

__global__ void gmm_prep_kernel(const float* __restrict__ w,
                                const float* __restrict__ loc,
                                const float* __restrict__ sp,
                                float* __restrict__ ws) {
    const int k = threadIdx.x;
    if (k >= GMM_K) return;

    float L[GMM_N][GMM_N];
    int t = 0;
    #pragma unroll
    for (int i = 0; i < GMM_N; ++i) {
        #pragma unroll
        for (int j = 0; j <= i; ++j) {
            L[i][j] = sp[t * GMM_K + k];
            ++t;
        }
    }

    // A = L^{-1} by forward substitution (lower triangular)
    float A[GMM_N][GMM_N];
    #pragma unroll
    for (int j = 0; j < GMM_N; ++j) {
        #pragma unroll
        for (int i = 0; i < GMM_N; ++i) A[i][j] = 0.0f;
        A[j][j] = 1.0f / L[j][j];
        #pragma unroll
        for (int i = 0; i < GMM_N; ++i) {
            if (i > j) {
                float s = 0.0f;
                #pragma unroll
                for (int m = 0; m < GMM_N; ++m)
                    if (m >= j && m < i) s += L[i][m] * A[m][j];
                A[i][j] = -s / L[i][i];
            }
        }
    }

    float logdet = 0.0f;
    #pragma unroll
    for (int i = 0; i < GMM_N; ++i) logdet += logf(L[i][i]);

    // log_softmax over the 4 weights
    float wv[GMM_K];
    #pragma unroll
    for (int kk = 0; kk < GMM_K; ++kk) wv[kk] = w[kk];
    float mx = wv[0];
    #pragma unroll
    for (int kk = 1; kk < GMM_K; ++kk) mx = fmaxf(mx, wv[kk]);
    float se = 0.0f;
    #pragma unroll
    for (int kk = 0; kk < GMM_K; ++kk) se += expf(wv[kk] - mx);
    const float logw = wv[k] - mx - logf(se);

    float* Minv = ws;
    float* cvec = ws + 256;
    float* ck   = ws + 288;

    #pragma unroll
    for (int n = 0; n < GMM_N; ++n) {
        #pragma unroll
        for (int c = 0; c < GMM_N; ++c)
            Minv[(k * GMM_N + n) * GMM_N + c] = A[n][c];
        float cv = 0.0f;
        #pragma unroll
        for (int c = 0; c < GMM_N; ++c)
            cv += A[n][c] * loc[c * GMM_K + k];   // mu_k[c] = loc[c][k]
        cvec[k * GMM_N + n] = cv;
    }
    ck[k] = logw - logdet - 0.5f * (float)GMM_N * LOG_2PI;
}

__global__ void __launch_bounds__(256)
gmm_main_kernel(const float* __restrict__ x,
                const float* __restrict__ ws,
                float* __restrict__ out,
                int numTiles) {
    const int lane = threadIdx.x & 31;
    const int hh   = lane >> 4;      // lane half: 0 or 1
    const int lm   = lane & 15;      // voxel-within-tile / row-within-halftile

    const float* __restrict__ Minv = ws;
    const float* __restrict__ cvec = ws + 256;
    const float* __restrict__ ck   = ws + 288;

    // --- A-matrix operands (V_WMMA_F32_16X16X4_F32 layout) ---
    // lane L: row M = L%16; VGPR0 carries K = 2*half, VGPR1 carries K = 2*half+1
    const int row0 = lm;        // output rows 0..15  -> clusters 0,1
    const int row1 = lm + 16;   // output rows 16..31 -> clusters 2,3
    const int ksel = 2 * hh;
    v2f a00 = { Minv[row0 * 8 + ksel    ], Minv[row0 * 8 + ksel + 1] };  // K 0..3
    v2f a01 = { Minv[row0 * 8 + 4 + ksel], Minv[row0 * 8 + 5 + ksel] };  // K 4..7
    v2f a10 = { Minv[row1 * 8 + ksel    ], Minv[row1 * 8 + ksel + 1] };
    v2f a11 = { Minv[row1 * 8 + 4 + ksel], Minv[row1 * 8 + 5 + ksel] };

    // --- C init = -c_m broadcast over columns (C layout: VGPR j = M j / j+8) ---
    v8f c0, c1;
    #pragma unroll
    for (int j = 0; j < 8; ++j) {
        c0[j] = -cvec[      8 * hh + j];
        c1[j] = -cvec[16 + 8 * hh + j];
    }
    const float cA = ck[hh];       // cluster 0/1 constant for this lane half
    const float cB = ck[2 + hh];   // cluster 2/3 constant

    const int waveId = (blockIdx.x * blockDim.x + threadIdx.x) >> 5;
    const int nWaves = (gridDim.x * blockDim.x) >> 5;

    for (int t = waveId; t < numTiles; t += nWaves) {
        // --- B-matrix operands: 2 consecutive channels of this lane's voxel ---
        const float* p = x + ((size_t)t * 16 + (size_t)lm) * GMM_N;
        v2f b0 = *(const v2f*)(p + ksel);       // channels {2h, 2h+1}
        v2f b1 = *(const v2f*)(p + 4 + ksel);   // channels {4+2h, 5+2h}

        // Z = M * X - c, accumulated over two K-chunks of 4
        v8f z0 = __builtin_amdgcn_wmma_f32_16x16x4_f32(
            false, a00, false, b0, (short)0, c0, false, false);
        z0 = __builtin_amdgcn_wmma_f32_16x16x4_f32(
            false, a01, false, b1, (short)0, z0, false, false);
        v8f z1 = __builtin_amdgcn_wmma_f32_16x16x4_f32(
            false, a10, false, b0, (short)0, c1, false, false);
        z1 = __builtin_amdgcn_wmma_f32_16x16x4_f32(
            false, a11, false, b1, (short)0, z1, false, false);

        // each lane holds the 8 z-values of one cluster for voxel lm
        float mahaA = 0.0f, mahaB = 0.0f;
        #pragma unroll
        for (int j = 0; j < 8; ++j) {
            mahaA = fmaf(z0[j], z0[j], mahaA);
            mahaB = fmaf(z1[j], z1[j], mahaB);
        }
        float lpA = fmaf(-0.5f, mahaA, cA);   // cluster hh
        float lpB = fmaf(-0.5f, mahaB, cB);   // cluster 2+hh

        // pair with the other lane-half's clusters for the same voxel
        float oA = __shfl_xor(lpA, 16, 32);
        float oB = __shfl_xor(lpB, 16, 32);

        float m = fmaxf(fmaxf(lpA, oA), fmaxf(lpB, oB));
        float s = __expf(lpA - m) + __expf(oA - m) +
                  __expf(lpB - m) + __expf(oB - m);
        float r = m + __logf(s);

        if (lane < 16) out[t * 16 + lm] = r;   // halves computed identical r
    }
}

extern "C" void kernel_launch(void* const* d_in, const int* in_sizes, int n_in,
                              void* d_out, int out_size, void* d_ws, size_t ws_size,
                              hipStream_t stream) {
    const float* x  = (const float*)d_in[0];   // [B, 8]
    const float* w  = (const float*)d_in[1];   // [4]
    const float* lc = (const float*)d_in[2];   // [8, 4]
    const float* sp = (const float*)d_in[3];   // [36, 4]
    float* out = (float*)d_out;
    float* ws  = (float*)d_ws;

    const int B = in_sizes[0] / GMM_N;
    const int numTiles = B / 16;               // B = 2^21 -> 131072 tiles

    gmm_prep_kernel<<<1, 32, 0, stream>>>(w, lc, sp, ws);

    const int threads = 256;                   // 8 waves / block
    int blocks = 2048;                         // 16384 waves, ~8 tiles each
    const int maxBlocks = (numTiles * 32 + threads - 1) / threads;
    if (blocks > maxBlocks) blocks = maxBlocks > 0 ? maxBlocks : 1;
    gmm_main_kernel<<<blocks, threads, 0, stream>>>(x, ws, out, numTiles);
}